// KTakesAll_26079041422027
// MI455X (gfx1250) — compile-verified
//
#include <hip/hip_runtime.h>
#include <stdint.h>

// KTakesAll: per row of N=2048 fp32, zero the k=1024 smallest entries
// (ties: lowest index first). One workgroup (256 threads, 8 waves @ wave32)
// per row. Row staged in LDS via gfx1250 async global->LDS loads; exact
// 4-pass MSB radix select with fully parallel (shfl + 2-level) scans.

#define N_COL    2048
#define NTHREADS 256
#define PER_T    8          // N_COL / NTHREADS
#define KSEL     1024       // int(N * 0.5)

#define GAS __attribute__((address_space(1)))
#define LAS __attribute__((address_space(3)))

typedef int v4i __attribute__((ext_vector_type(4)));

#ifndef __has_builtin
#define __has_builtin(x) 0
#endif

__device__ __forceinline__ unsigned order_key(float f) {
    // map float bits to unsigned so unsigned order == float ascending order
    unsigned u = __float_as_uint(f);
    return (u & 0x80000000u) ? ~u : (u | 0x80000000u);
}

__device__ __forceinline__ unsigned wave_incl_scan(unsigned v) {
    const unsigned lane = threadIdx.x & 31u;
    #pragma unroll
    for (int off = 1; off < 32; off <<= 1) {
        unsigned n = (unsigned)__shfl_up((int)v, off, 32);
        if (lane >= (unsigned)off) v += n;
    }
    return v;
}

// Inclusive scan across all 256 threads. Opens and closes with barriers
// (also makes back-to-back calls safe w.r.t. wsum reuse).
__device__ __forceinline__ unsigned block_incl_scan(unsigned v, unsigned* wsum) {
    const unsigned tid = threadIdx.x, lane = tid & 31u, wid = tid >> 5;
    __syncthreads();
    unsigned incl = wave_incl_scan(v);
    if (lane == 31u) wsum[wid] = incl;
    __syncthreads();
    if (wid == 0u) {
        unsigned w  = (lane < 8u) ? wsum[lane] : 0u;
        unsigned wi = wave_incl_scan(w);
        if (lane < 8u) wsum[lane] = wi - w;   // exclusive wave offsets
    }
    __syncthreads();
    return incl + wsum[wid];
}

__global__ __launch_bounds__(NTHREADS) void ktakesall_kernel(
        const float* __restrict__ g, float* __restrict__ out) {
    __shared__ float    rowBuf[N_COL];     // 8 KB row staging
    __shared__ unsigned hist[NTHREADS];    // 256-bin histogram
    __shared__ unsigned wsum[8];           // per-wave scan sums
    __shared__ unsigned s_bin, s_rem, s_tkey;

    const unsigned tid = threadIdx.x;
    const size_t   row = blockIdx.x;
    const float* gRow = g   + row * (size_t)N_COL;
    float*       oRow = out + row * (size_t)N_COL;

    // ---- stage row into LDS (async global->LDS path, ASYNCcnt) ----
#if __has_builtin(__builtin_amdgcn_global_load_async_to_lds_b128)
    #pragma unroll
    for (int j = 0; j < 2; ++j) {
        const float* src = gRow + (tid + j * NTHREADS) * 4;   // 16B per lane
        float*       dst = &rowBuf[(tid + j * NTHREADS) * 4];
        __builtin_amdgcn_global_load_async_to_lds_b128(
            (GAS v4i*)(uintptr_t)src, (LAS v4i*)(uintptr_t)dst, 0, 0);
    }
  #if __has_builtin(__builtin_amdgcn_s_wait_asynccnt)
    __builtin_amdgcn_s_wait_asynccnt(0);
  #else
    asm volatile("s_wait_asynccnt 0" ::: "memory");
  #endif
#else
    #pragma unroll
    for (int j = 0; j < 2; ++j)
        ((float4*)rowBuf)[tid + j * NTHREADS] =
            ((const float4*)gRow)[tid + j * NTHREADS];
#endif
    if (tid == 0) { s_rem = KSEL; s_tkey = 0u; }
    __syncthreads();

    // ---- load my 8 contiguous elements, build radix keys ----
    float    val[PER_T];
    unsigned key[PER_T];
    #pragma unroll
    for (int e = 0; e < PER_T; ++e) {
        val[e] = rowBuf[tid * PER_T + e];
        key[e] = order_key(val[e]);
    }

    // ---- 4-pass MSB-first radix select for the k-th smallest key ----
    unsigned matched = 0xFFu;  // bit e: element still matches the prefix
    #pragma unroll
    for (int pass = 0; pass < 4; ++pass) {
        const int shift = 24 - pass * 8;
        hist[tid] = 0u;
        __syncthreads();
        #pragma unroll
        for (int e = 0; e < PER_T; ++e)
            if (matched & (1u << e))
                atomicAdd(&hist[(key[e] >> shift) & 0xFFu], 1u);
        __syncthreads();
        const unsigned rem  = s_rem;          // prev-pass value (barriers order it)
        const unsigned cnt  = hist[tid];
        const unsigned incl = block_incl_scan(cnt, wsum);
        const unsigned excl = incl - cnt;
        if (excl < rem && rem <= incl) {      // unique winner bin == tid
            s_bin  = tid;
            s_rem  = rem - excl;              // rank remaining inside chosen bin
            s_tkey = (s_tkey << 8) | tid;
        }
        __syncthreads();
        const unsigned b = s_bin;
        #pragma unroll
        for (int e = 0; e < PER_T; ++e)
            if (((key[e] >> shift) & 0xFFu) != b) matched &= ~(1u << e);
    }

    const unsigned tkey      = s_tkey;  // exact k-th smallest key
    const unsigned numEqZero = s_rem;   // how many ties (== tkey) to zero

    // ---- index-ordered tie rank via block scan of equal-counts ----
    const unsigned eqCnt  = __popc(matched);  // matched now means key == tkey
    const unsigned inclEq = block_incl_scan(eqCnt, wsum);
    unsigned rank = inclEq - eqCnt;           // ties with smaller global index

    // ---- apply zeros: key < tkey always; first numEqZero ties too ----
    #pragma unroll
    for (int e = 0; e < PER_T; ++e) {
        const bool isEq = (matched >> e) & 1u;
        const bool z = (key[e] < tkey) || (isEq && rank < numEqZero);
        if (isEq) ++rank;
        if (z) val[e] = 0.0f;
    }

    // ---- coalesced b128 stores ----
    ((float4*)oRow)[tid * 2 + 0] = make_float4(val[0], val[1], val[2], val[3]);
    ((float4*)oRow)[tid * 2 + 1] = make_float4(val[4], val[5], val[6], val[7]);
}

extern "C" void kernel_launch(void* const* d_in, const int* in_sizes, int n_in,
                              void* d_out, int out_size, void* d_ws, size_t ws_size,
                              hipStream_t stream) {
    const float* g  = (const float*)d_in[0];
    float*      out = (float*)d_out;
    const int rows = in_sizes[0] / N_COL;     // 16 * 2048 = 32768
    hipLaunchKernelGGL(ktakesall_kernel, dim3(rows), dim3(NTHREADS), 0, stream,
                       g, out);
}